// PAFWeighter_23751169147304
// MI455X (gfx1250) — compile-verified
//
#include <hip/hip_runtime.h>

namespace {

constexpr int      KPTS  = 768;
constexpr int      NMAPS = 7;
constexpr unsigned MTOTU = 7u * 768u * 768u;   // 4,128,768 elements per plane

typedef __attribute__((ext_vector_type(2))) float v2f;
typedef __attribute__((ext_vector_type(8))) float v8f;

// Block = 256 threads = 8 wave32. Tile = 32 rows (p2) x 64 cols (p1).
// Each wave owns one 16x16 subtile; its R^2 matrix is computed by a single
// V_WMMA_F32_16X16X4_F32:  A-row(M) = (x2, y2, x2^2+y2^2, 1),
//                          B-col(N) = (-2*x1, -2*y1, 1, x1^2+y1^2)
// => D[M][N] = (x2-x1)^2 + (y2-y1)^2 exactly (all integers < 2^24, f32 RNE
//    accumulation is therefore exact -> R matches the reference's sqrt input).
__global__ __launch_bounds__(256)
void paf_weighter(const int* __restrict__ sk,
                  const float* __restrict__ paf,
                  float* __restrict__ out)
{
    __shared__ int2 rowXY[32];   // p2 coords of this block's 32 rows
    __shared__ int2 colXY[64];   // p1 coords of this block's 64 cols

    const int tid = threadIdx.x;
    const int bid = blockIdx.x;
    const int m      = bid / (24 * 12);
    const int rem    = bid % (24 * 12);
    const int rbase0 = (rem / 12) * 32;   // row-tile base within map
    const int cbase0 = (rem % 12) * 64;   // col-tile base within map

    // Stage tile coordinates into LDS. skeletons: int32 (seg, x, y) per point.
    if (tid < 32) {
        const int g = ((m + 1) * KPTS + rbase0 + tid) * 3;
        rowXY[tid] = make_int2(sk[g + 1], sk[g + 2]);
    } else if (tid < 96) {
        const int c = tid - 32;
        const int g = (m * KPTS + cbase0 + c) * 3;
        colXY[c] = make_int2(sk[g + 1], sk[g + 2]);
    }
    __syncthreads();

    const int  lane = tid & 31;
    const int  wave = tid >> 5;
    const int  ln   = lane & 15;
    const bool hi   = lane >= 16;
    const int  sr   = (wave >> 2) * 16;   // subtile row base within tile (0/16)
    const int  sc   = (wave & 3) * 16;    // subtile col base within tile (0..48)

    // This lane's output column (WMMA D layout: N = lane % 16).
    const int2 pc = colXY[sc + ln];
    const int  x1 = pc.x, y1 = pc.y;

#if __has_builtin(__builtin_amdgcn_wmma_f32_16x16x4_f32)
    // Build A (16x4 f32: lanes 0-15 -> K0,K1 in v0,v1; lanes 16-31 -> K2,K3)
    // and B (4x16 f32: same K split across lane halves, N = lane % 16).
    const int2  prA   = rowXY[sr + ln];
    const float nrm2r = (float)(prA.x * prA.x + prA.y * prA.y);
    const float nrm2c = (float)(x1 * x1 + y1 * y1);
    v2f A, B;
    A[0] = hi ? nrm2r : (float)prA.x;        // K2 : K0
    A[1] = hi ? 1.0f  : (float)prA.y;        // K3 : K1
    B[0] = hi ? 1.0f  : (float)(-2 * x1);    // K2 : K0
    B[1] = hi ? nrm2c : (float)(-2 * y1);    // K3 : K1
    v8f acc = {};
    const v8f R2v = __builtin_amdgcn_wmma_f32_16x16x4_f32(
        /*neg_a=*/false, A, /*neg_b=*/false, B,
        /*c_mod=*/(short)0, acc, /*reuse_a=*/false, /*reuse_b=*/false);
#endif

    const float* __restrict__ paf0 = paf + (size_t)m * (2u * 1024u * 1024u);
    const float* __restrict__ paf1 = paf0 + 1024u * 1024u;

    // Uniform per-plane base pointers (SALU once) + shared 32-bit offset per
    // store: keeps all four NT stores on SGPR-base + 32-bit-VGPR addressing,
    // eliminating the per-pair 64-bit VALU index chains.
    float* __restrict__ o0 = out;              // i1 ids
    float* __restrict__ o1 = out + MTOTU;      // i2 ids
    float* __restrict__ o2 = out + 2 * MTOTU;  // li
    float* __restrict__ o3 = out + 3 * MTOTU;  // R

    const int      cc      = cbase0 + sc + ln;        // column within map
    const float    i1f     = (float)(m * KPTS + cc);  // p1 id (exact in f32)
    const unsigned mapBase = (unsigned)m * (768u * 768u);

    #pragma unroll
    for (int v = 0; v < 8; ++v) {
        const int  M  = v + (hi ? 8 : 0);    // WMMA D layout: M = vgpr + 8*hi
        const int2 p2 = rowXY[sr + M];
        const int  x2 = p2.x, y2 = p2.y;

        // 10-point integer line gather; PAF layout [m][ch][y][x], 1024x1024.
        // Unconditional + branchless epilogue so the scheduler can interleave
        // the 160 gathers of the fully-unrolled v-loop for MLP.
        float s0 = 0.0f, s1 = 0.0f;
        #pragma unroll
        for (int t = 0; t < 10; ++t) {
            const unsigned lx  = (unsigned)(x1 * (9 - t) + x2 * t) / 9u;
            const unsigned ly  = (unsigned)(y1 * (9 - t) + y2 * t) / 9u;
            const unsigned off = (ly << 10) | lx;
            s0 += paf0[off];
            s1 += paf1[off];
        }

#if __has_builtin(__builtin_amdgcn_wmma_f32_16x16x4_f32)
        const float r2 = R2v[v];
#else
        const float dxf0 = (float)(x2 - x1), dyf0 = (float)(y2 - y1);
        const float r2   = dxf0 * dxf0 + dyf0 * dyf0;
#endif
        // Raw 1-ulp hardware sqrt/rsq (input is an exact integer in
        // [0, 2.1e6] -- no denormal/overflow fixups needed; ~1.2e-7 rel
        // error, same order as the li sum re-association already present).
#if __has_builtin(__builtin_amdgcn_sqrtf) && __has_builtin(__builtin_amdgcn_rsqf)
        const float R    = __builtin_amdgcn_sqrtf(r2);
        const float rinv = __builtin_amdgcn_rsqf(r2);
#else
        const float R    = sqrtf(r2);
        const float rinv = 1.0f / R;
#endif
        // Branchless R==0 handling (reference maps the resulting NaN to 0):
        // s0/s1 are always finite, so li == 0 exactly when inv == 0.
        const float inv = (r2 > 0.0f) ? (0.1f * rinv) : 0.0f;   // v_cndmask
        const float li  = ((float)(x2 - x1) * s0 + (float)(y2 - y1) * s1) * inv;

        const int      rr  = rbase0 + sr + M;          // row within map
        const unsigned idx = mapBase + (unsigned)rr * 768u + (unsigned)cc;
        // Output is 66 MB write-once: NT stores keep L2 free for PAF gathers.
        __builtin_nontemporal_store(i1f,                          o0 + idx);
        __builtin_nontemporal_store((float)((m + 1) * KPTS + rr), o1 + idx);
        __builtin_nontemporal_store(li,                           o2 + idx);
        __builtin_nontemporal_store(R,                            o3 + idx);
    }
}

} // namespace

extern "C" void kernel_launch(void* const* d_in, const int* in_sizes, int n_in,
                              void* d_out, int out_size, void* d_ws, size_t ws_size,
                              hipStream_t stream)
{
    (void)in_sizes; (void)n_in; (void)out_size; (void)d_ws; (void)ws_size;
    const int*   sk  = (const int*)d_in[0];     // skeletons: (6144, 3) int32
    const float* paf = (const float*)d_in[1];   // PAF: (7, 2, 1024, 1024) f32
    float*       out = (float*)d_out;           // 4 planes x 4,128,768 f32

    paf_weighter<<<dim3(NMAPS * 24 * 12), dim3(256), 0, stream>>>(sk, paf, out);
}